// Pool_83631603187831
// MI455X (gfx1250) — compile-verified
//
#include <hip/hip_runtime.h>
#include <stdint.h>

typedef float v2f __attribute__((ext_vector_type(2)));
typedef float v8f __attribute__((ext_vector_type(8)));

// ---------------------------------------------------------------------------
// Kernel 1: scores via V_WMMA_F32_16X16X4_F32.
// One wave32 handles 16 rows of x. A = 16x4 tile of x, B = w broadcast into
// all 16 columns -> D[:,j] identical for all j = the 16 dot products.
// Writes radix-sortable keys: ascending key order == descending h order,
// stable radix then matches jax.lax.top_k tie-breaking (lower index first).
// ---------------------------------------------------------------------------
__global__ void scores_wmma_kernel(const float* __restrict__ x,
                                   const float* __restrict__ w,
                                   unsigned* __restrict__ keys,
                                   int N, int F) {
    const int lane      = threadIdx.x & 31;
    const int waveInBlk = threadIdx.x >> 5;
    const int wave      = blockIdx.x * (blockDim.x >> 5) + waveInBlk;
    const long rowBase  = (long)wave * 16;
    if (rowBase >= N) return;                   // uniform across the wave

    const int  rowInWave = lane & 15;
    const int  kHalf     = lane >> 4;           // lanes 0-15: K pair 0,1 ; 16-31: K pair 2,3
    long row  = rowBase + rowInWave;
    long rowC = row < N ? row : (long)(N - 1);  // clamp (select, EXEC untouched)
    const float* xr = x + rowC * (long)F;

    v8f c = {};
    for (int k0 = 0; k0 < F; k0 += 4) {
        const int kk = k0 + 2 * kHalf;
        v2f a;  a.x  = xr[kk]; a.y  = xr[kk + 1];
        v2f bv; bv.x = w[kk];  bv.y = w[kk + 1];
        c = __builtin_amdgcn_wmma_f32_16x16x4_f32(
                false, a, false, bv, (short)0, c, false, false);
    }

    // D layout: VGPR r -> lanes 0-15: (M=r, N=lane); lanes 16-31: (M=8+r, N=lane-16)
    // Column N=0 lives in lane 0 (M=0..7) and lane 16 (M=8..15).
    if ((lane & 15) == 0) {
        for (int r = 0; r < 8; ++r) {
            long m = rowBase + (long)kHalf * 8 + r;
            if (m < N) {
                float h = c[r];                 // bias omitted: rank-invariant
                unsigned bits = __float_as_uint(h);
                unsigned mono = (bits & 0x80000000u) ? ~bits : (bits | 0x80000000u);
                keys[m] = ~mono;                // ascending key == descending h
            }
        }
    }
}

// ---------------------------------------------------------------------------
// Init: idx[n]=n, inv[n]=-1 (n<N), deg[r]=0 (r<k)
// ---------------------------------------------------------------------------
__global__ void init_kernel(unsigned* __restrict__ idx, int* __restrict__ inv,
                            float* __restrict__ deg, int N, int k) {
    int i = blockIdx.x * blockDim.x + threadIdx.x;
    if (i < N) { idx[i] = (unsigned)i; inv[i] = -1; }
    if (i < k) { deg[i] = 0.0f; }
}

// ---------------------------------------------------------------------------
// Radix sort (8-bit digits, 4 passes, stable)
// ---------------------------------------------------------------------------
__global__ void radix_hist_kernel(const unsigned* __restrict__ keys, int n, int shift,
                                  unsigned* __restrict__ blockHist, int numBlocks) {
    __shared__ unsigned sh[256];
    const int tid = threadIdx.x;
    sh[tid] = 0;
    __syncthreads();
    int i = blockIdx.x * 256 + tid;
    if (i < n) {
        unsigned bin = (keys[i] >> shift) & 255u;
        atomicAdd(&sh[bin], 1u);
    }
    __syncthreads();
    blockHist[(size_t)tid * numBlocks + blockIdx.x] = sh[tid];
}

__global__ void radix_scan_kernel(unsigned* __restrict__ blockHist, int numBlocks) {
    __shared__ unsigned tot[256];
    __shared__ unsigned binStart[256];
    const int b = threadIdx.x;
    unsigned* p = blockHist + (size_t)b * numBlocks;
    unsigned run = 0;
    for (int i = 0; i < numBlocks; ++i) { unsigned t = p[i]; p[i] = run; run += t; }
    tot[b] = run;
    __syncthreads();
    if (b == 0) {
        unsigned s = 0;
        for (int i = 0; i < 256; ++i) { binStart[i] = s; s += tot[i]; }
    }
    __syncthreads();
    const unsigned bs = binStart[b];
    for (int i = 0; i < numBlocks; ++i) p[i] += bs;
}

__global__ void radix_scatter_kernel(const unsigned* __restrict__ keysIn,
                                     const unsigned* __restrict__ idxIn,
                                     unsigned* __restrict__ keysOut,
                                     unsigned* __restrict__ idxOut,
                                     int n, int shift,
                                     const unsigned* __restrict__ blockHist,
                                     int numBlocks) {
    __shared__ int sbins[256];
    const int tid = threadIdx.x;
    int i = blockIdx.x * 256 + tid;
    unsigned key = 0, idx = 0;
    int bin = 256;                               // sentinel for inactive lanes
    if (i < n) {
        key = keysIn[i];
        idx = idxIn[i];
        bin = (int)((key >> shift) & 255u);
    }
    sbins[tid] = bin;
    __syncthreads();
    if (i < n) {
        int rank = 0;
        for (int j = 0; j < tid; ++j) rank += (sbins[j] == bin) ? 1 : 0;
        unsigned dst = blockHist[(size_t)bin * numBlocks + blockIdx.x] + (unsigned)rank;
        keysOut[dst] = key;
        idxOut[dst]  = idx;
    }
}

// ---------------------------------------------------------------------------
// Post-sort: inv[topIdx[r]] = r, indices output (as float)
// ---------------------------------------------------------------------------
__global__ void postsort_kernel(const unsigned* __restrict__ sortedIdx,
                                int k, int* __restrict__ inv,
                                float* __restrict__ outIndices) {
    int r = blockIdx.x * blockDim.x + threadIdx.x;
    if (r < k) {
        unsigned id = sortedIdx[r];
        inv[id] = r;
        outIndices[r] = (float)id;
    }
}

// new_x[r,f] = x[idx[r], f]  (128-bit vectorized; F % 4 == 0)
__global__ void gather_x_kernel(const float* __restrict__ x,
                                const unsigned* __restrict__ sortedIdx,
                                float* __restrict__ outNewX,
                                int k, int F4) {
    long i = (long)blockIdx.x * blockDim.x + threadIdx.x;
    long total = (long)k * F4;
    if (i < total) {
        long r = i / F4;
        long f = i - r * F4;
        const float4* xv = reinterpret_cast<const float4*>(x);
        float4* ov = reinterpret_cast<float4*>(outNewX);
        ov[i] = xv[(long)sortedIdx[r] * F4 + f];
    }
}

// ---------------------------------------------------------------------------
// Edge relabel + mask + degree accumulation (segment_sum via f32 atomics)
// ---------------------------------------------------------------------------
__global__ void edges_kernel(const long long* __restrict__ ei,
                             const int* __restrict__ inv,
                             float* __restrict__ outEdges,   // [2E]
                             float* __restrict__ outMask,    // [E]
                             int* __restrict__ srcArr,       // [E] (-1 if masked)
                             float* __restrict__ deg,        // [k]
                             long E) {
    long e = (long)blockIdx.x * blockDim.x + threadIdx.x;
    if (e < E) {
        long long s0 = ei[e];
        long long d0 = ei[E + e];
        int s = inv[s0];
        int d = inv[d0];
        bool m = (s >= 0) && (d >= 0);
        int sw = m ? s : 0;
        int dw = m ? d : 0;
        outEdges[e]     = (float)sw;
        outEdges[E + e] = (float)dw;
        outMask[e]      = m ? 1.0f : 0.0f;
        srcArr[e]       = m ? s : -1;
        if (m) atomicAdd(&deg[s], 1.0f);
    }
}

__global__ void recip_kernel(float* __restrict__ deg, int k) {
    int r = blockIdx.x * blockDim.x + threadIdx.x;
    if (r < k) {
        float v = deg[r];
        deg[r] = (v > 0.0f) ? (1.0f / v) : 0.0f;
    }
}

__global__ void attr_kernel(const int* __restrict__ srcArr,
                            const float* __restrict__ degInv,
                            float* __restrict__ outAttr, long E) {
    long e = (long)blockIdx.x * blockDim.x + threadIdx.x;
    if (e < E) {
        int s = srcArr[e];
        outAttr[e] = (s >= 0) ? degInv[s] : 0.0f;
    }
}

// ---------------------------------------------------------------------------
extern "C" void kernel_launch(void* const* d_in, const int* in_sizes, int n_in,
                              void* d_out, int out_size, void* d_ws, size_t ws_size,
                              hipStream_t stream) {
    const float*     x  = (const float*)d_in[0];
    const long long* ei = (const long long*)d_in[1];
    const float*     w  = (const float*)d_in[2];
    // d_in[3] (b_proj) is rank-invariant and new_x uses raw x: unused.

    const int F = in_sizes[2];          // w_proj flat size == F
    const int N = in_sizes[0] / F;
    const long E = (long)in_sizes[1] / 2;
    int k = (int)(0.5 * (double)N);
    if (k < 2) k = 2;

    // Output layout (all float32, return order):
    float* out       = (float*)d_out;
    float* outNewX   = out;
    float* outEdges  = outNewX + (size_t)k * F;
    float* outAttr   = outEdges + (size_t)2 * E;
    float* outMask   = outAttr + (size_t)E;
    float* outIdx    = outMask + (size_t)E;

    // Workspace carve-up
    const int sortBlocks = (N + 255) / 256;
    char* ws = (char*)d_ws;
    size_t off = 0;
    auto carve = [&](size_t bytes) -> void* {
        void* p = ws + off;
        off = (off + bytes + 255) & ~(size_t)255;
        return p;
    };
    unsigned* keysA     = (unsigned*)carve((size_t)N * 4);
    unsigned* keysB     = (unsigned*)carve((size_t)N * 4);
    unsigned* idxA      = (unsigned*)carve((size_t)N * 4);
    unsigned* idxB      = (unsigned*)carve((size_t)N * 4);
    unsigned* blockHist = (unsigned*)carve((size_t)256 * sortBlocks * 4);
    int*      inv       = (int*)carve((size_t)N * 4);
    float*    deg       = (float*)carve((size_t)k * 4);
    int*      srcArr    = (int*)carve((size_t)E * 4);
    (void)ws_size; (void)n_in; (void)out_size;

    // 1) scores -> sort keys (WMMA f32 16x16x4)
    {
        int waves  = (N + 15) / 16;
        int blocks = (waves + 7) / 8;           // 8 waves per 256-thread block
        scores_wmma_kernel<<<blocks, 256, 0, stream>>>(x, w, keysA, N, F);
    }

    // 2) init idx / inv / deg
    {
        int m = N > k ? N : k;
        init_kernel<<<(m + 255) / 256, 256, 0, stream>>>(idxA, inv, deg, N, k);
    }

    // 3) stable LSD radix sort (4 x 8-bit), ascending keys == descending scores
    {
        unsigned* kin = keysA; unsigned* vin = idxA;
        unsigned* kot = keysB; unsigned* vot = idxB;
        for (int pass = 0; pass < 4; ++pass) {
            int shift = pass * 8;
            radix_hist_kernel<<<sortBlocks, 256, 0, stream>>>(kin, N, shift, blockHist, sortBlocks);
            radix_scan_kernel<<<1, 256, 0, stream>>>(blockHist, sortBlocks);
            radix_scatter_kernel<<<sortBlocks, 256, 0, stream>>>(kin, vin, kot, vot,
                                                                 N, shift, blockHist, sortBlocks);
            unsigned* t;
            t = kin; kin = kot; kot = t;
            t = vin; vin = vot; vot = t;
        }
        // 4 passes: final sorted sequence is back in keysA/idxA
    }

    // 4) inv table + indices output
    postsort_kernel<<<(k + 255) / 256, 256, 0, stream>>>(idxA, k, inv, outIdx);

    // 5) new_x gather (float4)
    {
        int F4 = F / 4;
        long total = (long)k * F4;
        gather_x_kernel<<<(int)((total + 255) / 256), 256, 0, stream>>>(x, idxA, outNewX, k, F4);
    }

    // 6) edge relabel + mask + degree
    edges_kernel<<<(int)((E + 255) / 256), 256, 0, stream>>>(ei, inv, outEdges, outMask,
                                                             srcArr, deg, E);

    // 7) deg -> deg_inv (in place)
    recip_kernel<<<(k + 255) / 256, 256, 0, stream>>>(deg, k);

    // 8) edge_attr = deg_inv[src] * ew
    attr_kernel<<<(int)((E + 255) / 256), 256, 0, stream>>>(srcArr, deg, outAttr, E);
}